// NodewiseMassBias_90958817395266
// MI455X (gfx1250) — compile-verified
//
#include <hip/hip_runtime.h>
#include <math.h>

#define B_ 16
#define T_ 2048
#define D_ 512
#define H_ 64

typedef float v2f __attribute__((ext_vector_type(2)));
typedef float v8f __attribute__((ext_vector_type(8)));

// ---------------- Kernel 1: build P = [E, Pt*cos(phi), Pt*sin(phi), Pt*sinh(clip(eta))] ----------------
__global__ void build_p_kernel(const float* __restrict__ tok, float4* __restrict__ P, int n) {
  int i = blockIdx.x * blockDim.x + threadIdx.x;
  if (i >= n) return;
  float4 t = ((const float4*)tok)[i];
  float E = t.x, Pt = t.y, eta = t.z, phi = t.w;
  float c = cosf(phi), s = sinf(phi);
  float e = fminf(fmaxf(eta, -20.f), 20.f);
  float4 p;
  p.x = E; p.y = Pt * c; p.z = Pt * s; p.w = Pt * sinhf(e);
  P[i] = p;
}

// ---------------- top-8 helpers ----------------
template <int MASK>
__device__ __forceinline__ float sxf(float v) {
  return __int_as_float(__builtin_amdgcn_ds_swizzle(__float_as_int(v), (MASK << 10) | 0x1f));
}
template <int MASK>
__device__ __forceinline__ int sxi(int v) {
  return __builtin_amdgcn_ds_swizzle(v, (MASK << 10) | 0x1f);
}
// ordering: (value desc, index asc) — matches lax.top_k tie-breaking
__device__ __forceinline__ bool gtp(float a, int ai, float b, int bi) {
  return (a > b) || (a == b && ai < bi);
}
__device__ __forceinline__ void ce(float& av, int& ai, float& bv, int& bi) {
  bool sw = gtp(bv, bi, av, ai);
  float t1 = sw ? bv : av; int t2 = sw ? bi : ai;
  float t3 = sw ? av : bv; int t4 = sw ? ai : bi;
  av = t1; ai = t2; bv = t3; bi = t4;
}
// merge my sorted-desc 8 with XOR-partner's sorted-desc 8 -> top-8 sorted desc (same in both lanes)
template <int MASK>
__device__ __forceinline__ void mergeStep(float v[8], int ix[8]) {
  float ov[8]; int oi[8];
#pragma unroll
  for (int i = 0; i < 8; i++) { ov[i] = sxf<MASK>(v[i]); oi[i] = sxi<MASK>(ix[i]); }
  float c[8]; int ci[8];
#pragma unroll
  for (int i = 0; i < 8; i++) {  // bitonic first stage: top-8 of union, bitonic order
    bool g = gtp(v[i], ix[i], ov[7 - i], oi[7 - i]);
    c[i] = g ? v[i] : ov[7 - i];
    ci[i] = g ? ix[i] : oi[7 - i];
  }
  // bitonic merge (desc) of bitonic sequence of 8: distances 4,2,1
  ce(c[0],ci[0],c[4],ci[4]); ce(c[1],ci[1],c[5],ci[5]); ce(c[2],ci[2],c[6],ci[6]); ce(c[3],ci[3],c[7],ci[7]);
  ce(c[0],ci[0],c[2],ci[2]); ce(c[1],ci[1],c[3],ci[3]); ce(c[4],ci[4],c[6],ci[6]); ce(c[5],ci[5],c[7],ci[7]);
  ce(c[0],ci[0],c[1],ci[1]); ce(c[2],ci[2],c[3],ci[3]); ce(c[4],ci[4],c[5],ci[5]); ce(c[6],ci[6],c[7],ci[7]);
#pragma unroll
  for (int i = 0; i < 8; i++) { v[i] = c[i]; ix[i] = ci[i]; }
}

// ---------------- Kernel 2: one wave32 per row: Minkowski dots, masked top-8, masses ----------------
__global__ void topk_mass_kernel(const float4* __restrict__ P,
                                 const unsigned char* __restrict__ mask,
                                 float* __restrict__ massWS) {
  int wave = threadIdx.x >> 5;
  int lane = threadIdx.x & 31;
  int row = blockIdx.x * 8 + wave;   // [0, B*T)
  int b = row >> 11;                 // row / T_
  int t = row & (T_ - 1);
  float4 pt = P[row];
  float pmE = pt.x, pmx = -pt.y, pmy = -pt.z, pmz = -pt.w;

  float vals[8]; int inds[8];
#pragma unroll
  for (int i = 0; i < 8; i++) { vals[i] = -INFINITY; inds[i] = 0x7FFFFFFF; }

  const float4* Pb = P + b * T_;
  const unsigned char* mb = mask + b * T_;

  for (int s = lane; s < T_; s += 32) {
    float4 ps = Pb[s];
    float dot = pmE * ps.x + pmx * ps.y + pmy * ps.z + pmz * ps.w;
    float v = mb[s] ? dot : -INFINITY;
    if (gtp(v, s, vals[7], inds[7])) {      // rare-taken guard
#pragma unroll
      for (int i = 7; i >= 0; i--) {        // branch-free sorted insert
        bool gi = gtp(v, s, vals[i], inds[i]);
        bool gm = (i > 0) ? gtp(v, s, vals[i - 1], inds[i - 1]) : false;
        float nv = gi ? (gm ? vals[i - 1] : v) : vals[i];
        int   ni = gi ? (gm ? inds[i - 1] : s) : inds[i];
        vals[i] = nv; inds[i] = ni;
      }
    }
  }
  mergeStep<1>(vals, inds);
  mergeStep<2>(vals, inds);
  mergeStep<4>(vals, inds);
  mergeStep<8>(vals, inds);
  mergeStep<16>(vals, inds);

  if (lane == 0) {
    float maskf = (float)mb[t];
    float cE = 0.f, cx = 0.f, cy = 0.f, cz = 0.f;
    float m[3];
#pragma unroll
    for (int i = 0; i < 8; i++) {
      float4 p = Pb[inds[i]];
      cE += p.x; cx += p.y; cy += p.z; cz += p.w;
      if (i == 1 || i == 3 || i == 7) {
        float m2 = cE * cE - (cx * cx + cy * cy + cz * cz);
        m2 = fmaxf(m2, 0.f) * maskf;
        float mm = sqrtf(m2 + 1e-8f);
        if (i == 1) m[0] = mm; else if (i == 3) m[1] = mm; else m[2] = mm;
      }
    }
    massWS[row * 3 + 0] = m[0];
    massWS[row * 3 + 1] = m[1];
    massWS[row * 3 + 2] = m[2];
  }
}

// ---------------- Kernel 3: h = gelu(mass@W1 + b1); out = h@W2 + b2 via V_WMMA_F32_16X16X4_F32 ----------------
// Block = 128 threads (4 waves) handles 16 rows x 512 cols; each wave owns 128 cols (8 N-tiles).
__global__ void mlp_kernel(const float* __restrict__ massWS,
                           const float* __restrict__ W1,
                           const float* __restrict__ b1,
                           const float* __restrict__ W2,
                           const float* __restrict__ b2,
                           float* __restrict__ out) {
  __shared__ float h_lds[16 * 64];
  int row0 = blockIdx.x * 16;
  int tid = threadIdx.x;  // 0..127
  {
    int m = tid & 15;
    int j0 = (tid >> 4) * 8;
    int row = row0 + m;
    float m0 = massWS[row * 3 + 0];
    float m1 = massWS[row * 3 + 1];
    float m2v = massWS[row * 3 + 2];
#pragma unroll
    for (int jj = 0; jj < 8; jj++) {
      int j = j0 + jj;
      float pre = m0 * W1[j] + m1 * W1[64 + j] + m2v * W1[128 + j] + b1[j];
      float g = 0.5f * pre * (1.0f + erff(pre * 0.70710678118654752f));  // exact gelu
      h_lds[m * 64 + j] = g;
    }
  }
  __syncthreads();

  int w = tid >> 5;
  int lane = tid & 31;
  int half = lane >> 4;   // lanes 16-31 carry K+2 (A) / M+8 (D)
  int mrow = lane & 15;

  // A operand (16x4 f32, 2 VGPRs): lane holds h[mrow][4*kk + 2*half + {0,1}] for all 16 K-chunks
  v2f a[16];
  const v2f* hl2 = (const v2f*)(&h_lds[mrow * 64]);
#pragma unroll
  for (int kk = 0; kk < 16; kk++) a[kk] = hl2[2 * kk + half];

#pragma unroll 1
  for (int nt = 0; nt < 8; nt++) {
    int n0 = w * 128 + nt * 16;
    int col = n0 + mrow;
    v8f acc = {};
#pragma unroll
    for (int kk = 0; kk < 16; kk++) {
      int kb = 4 * kk + 2 * half;
      v2f bb;
      bb.x = W2[kb * 512 + col];
      bb.y = W2[(kb + 1) * 512 + col];
      acc = __builtin_amdgcn_wmma_f32_16x16x4_f32(false, a[kk], false, bb,
                                                  (short)0, acc, false, false);
    }
    float bias = b2[col];
#pragma unroll
    for (int r = 0; r < 8; r++) {   // D layout: lanes0-15 -> M=r, lanes16-31 -> M=8+r
      int row = row0 + r + 8 * half;
      out[(size_t)row * 512 + col] = acc[r] + bias;
    }
  }
}

extern "C" void kernel_launch(void* const* d_in, const int* in_sizes, int n_in,
                              void* d_out, int out_size, void* d_ws, size_t ws_size,
                              hipStream_t stream) {
  const float* tok = (const float*)d_in[0];
  const unsigned char* mask = (const unsigned char*)d_in[1];
  const float* W1 = (const float*)d_in[2];
  const float* b1 = (const float*)d_in[3];
  const float* W2 = (const float*)d_in[4];
  const float* b2 = (const float*)d_in[5];
  float* out = (float*)d_out;

  float4* P = (float4*)d_ws;                                        // B*T float4 = 512 KB
  float* massWS = (float*)((char*)d_ws + (size_t)B_ * T_ * 4 * 4);  // B*T*3 floats

  int n = B_ * T_;
  build_p_kernel<<<(n + 255) / 256, 256, 0, stream>>>(tok, P, n);
  topk_mass_kernel<<<n / 8, 256, 0, stream>>>(P, mask, massWS);
  mlp_kernel<<<n / 16, 128, 0, stream>>>(massWS, W1, b1, W2, b2, out);
}